// LocalWLNet_7164005450027
// MI455X (gfx1250) — compile-verified
//
#include <hip/hip_runtime.h>
#include <math.h>

#define EPSV 1e-5f

typedef __attribute__((ext_vector_type(2))) float v2f;
typedef __attribute__((ext_vector_type(8))) float v8f;

// ---------------------------------------------------------------- utilities

__global__ void fill4_kernel(float4* __restrict__ p, float v, long n4) {
  long i = (long)blockIdx.x * blockDim.x + threadIdx.x;
  long stride = (long)gridDim.x * blockDim.x;
  float4 val = make_float4(v, v, v, v);
  for (; i < n4; i += stride) p[i] = val;
}

// out rows of C1=32 floats = 8 float4 per row
__global__ void gather_emb_kernel(const float4* __restrict__ emb, const int* __restrict__ ids,
                                  float4* __restrict__ out, long total4) {
  long i = (long)blockIdx.x * blockDim.x + threadIdx.x;
  long stride = (long)gridDim.x * blockDim.x;
  for (; i < total4; i += stride) {
    long r = i >> 3;           // row (C=32 -> 8 groups)
    int g = (int)(i & 7);
    out[i] = emb[(long)ids[r] * 8 + g];
  }
}

__global__ void deg_kernel(const int* __restrict__ dst, float* __restrict__ deg, long E) {
  long i = (long)blockIdx.x * blockDim.x + threadIdx.x;
  long stride = (long)gridDim.x * blockDim.x;
  for (; i < E; i += stride) atomicAdd(&deg[dst[i]], 1.0f);
}

__global__ void rsqrt_kernel(float* __restrict__ p, long n) {
  long i = (long)blockIdx.x * blockDim.x + threadIdx.x;
  long stride = (long)gridDim.x * blockDim.x;
  for (; i < n; i += stride) {
    float d = p[i];
    p[i] = (d > 0.0f) ? rsqrtf(d) : 0.0f;
  }
}

// ------------------------------------------------------------- GraphNorm

// pass 0: out[c] += sum_i x[i,c]
// pass 1: out[c] += sum_i (x[i,c]-a[c]*mean[c])^2, mean = meansum*invN
template <int C>
__global__ void colsum_kernel(const float4* __restrict__ x, const float* __restrict__ a,
                              const float* __restrict__ meansum, float* __restrict__ out,
                              long total4, float invN, int pass) {
  constexpr int CQ = C / 4;
  __shared__ float sh[32];
  if (threadIdx.x < 32) sh[threadIdx.x] = 0.0f;
  __syncthreads();
  long i = (long)blockIdx.x * blockDim.x + threadIdx.x;
  long stride = (long)gridDim.x * blockDim.x;
  for (; i < total4; i += stride) {
    int c0 = (int)(i % CQ) * 4;
    float4 v = x[i];
    float e[4] = {v.x, v.y, v.z, v.w};
#pragma unroll
    for (int j = 0; j < 4; ++j) {
      float t = e[j];
      if (pass) {
        float d = t - a[c0 + j] * (meansum[c0 + j] * invN);
        t = d * d;
      }
      atomicAdd(&sh[c0 + j], t);   // ds_add_f32
    }
  }
  __syncthreads();
  if ((int)threadIdx.x < C) atomicAdd(&out[threadIdx.x], sh[threadIdx.x]);
}

template <int C>
__global__ void gnorm_apply_kernel(float4* __restrict__ x, const float* __restrict__ colsum,
                                   const float* __restrict__ colsq, const float* __restrict__ w,
                                   const float* __restrict__ b, const float* __restrict__ a,
                                   long total4, float invN, int relu) {
  constexpr int CQ = C / 4;
  long i = (long)blockIdx.x * blockDim.x + threadIdx.x;
  long stride = (long)gridDim.x * blockDim.x;
  for (; i < total4; i += stride) {
    int c0 = (int)(i % CQ) * 4;
    float4 v = x[i];
    float e[4] = {v.x, v.y, v.z, v.w};
#pragma unroll
    for (int j = 0; j < 4; ++j) {
      int c = c0 + j;
      float mean = colsum[c] * invN;
      float var  = colsq[c] * invN;
      float t = w[c] * (e[j] - a[c] * mean) * rsqrtf(var + EPSV) + b[c];
      if (relu) t = fmaxf(t, 0.0f);
      e[j] = t;
    }
    x[i] = make_float4(e[0], e[1], e[2], e[3]);
  }
}

// ------------------------------------------------------------- WMMA GEMM
// H[Mpad,COUT] = X[Mpad,CIN] @ W[CIN,COUT]; Mpad multiple of 64 (padded workspace),
// so there are NO bounds checks -> branch-free vector loads feeding
// V_WMMA_F32_16X16X4_F32. Block: 8 waves -> 64 rows x 32 cols.

template <int CIN, int COUT>
__global__ void gemm_wmma_kernel(const float* __restrict__ X, const float* __restrict__ W,
                                 float* __restrict__ H) {
  __shared__ float Wl[32 * 32];
  for (int i = threadIdx.x; i < 32 * 32; i += 256) {
    int r = i >> 5, c = i & 31;
    Wl[i] = (r < CIN && c < COUT) ? W[r * COUT + c] : 0.0f;
  }
  __syncthreads();

  int wave = threadIdx.x >> 5;   // 0..7
  int lane = threadIdx.x & 31;
  long rowBase = (long)blockIdx.x * 64 + (wave >> 1) * 16;
  int colBase = (wave & 1) * 16;

  int half = lane >> 4;          // selects K pair (0,1) vs (2,3)
  int l16  = lane & 15;
  const float* Xr = X + (rowBase + l16) * CIN + half * 2;  // 8B-aligned pairs
  int bcol = colBase + l16;

  v8f acc = {};
#pragma unroll
  for (int k0 = 0; k0 < CIN; k0 += 4) {
    v2f afrag = *(const v2f*)(Xr + k0);          // single b64 load
    int ka = k0 + half * 2;
    v2f bfrag;
    bfrag.x = Wl[(ka    ) * 32 + bcol];
    bfrag.y = Wl[(ka + 1) * 32 + bcol];
    acc = __builtin_amdgcn_wmma_f32_16x16x4_f32(
        false, afrag, false, bfrag, (short)0, acc, false, false);
  }

#pragma unroll
  for (int i = 0; i < 8; ++i) {
    long r = rowBase + i + half * 8;
    int  c = colBase + l16;
    if (COUT == 32 || c < COUT) H[r * COUT + c] = acc[i];
  }
}

// ------------------------------------------------------------- GCN scatter

// agg[d,:] += h[s,:] * dinv[s]*dinv[d]; one float4 channel-group per thread.
// Fixed grid + grid-stride so each thread iterates; prefetch next gather row.
template <int C>
__global__ void scatter_kernel(const float* __restrict__ h, const int* __restrict__ src,
                               const int* __restrict__ dst, const float* __restrict__ dinv,
                               float* __restrict__ agg, long E) {
  constexpr int GROUPS = C / 4;
  long total = E * GROUPS;
  long t = (long)blockIdx.x * blockDim.x + threadIdx.x;
  long stride = (long)gridDim.x * blockDim.x;
  for (; t < total; t += stride) {
    long e = t / GROUPS;
    int g = (int)(t % GROUPS);
    int s = src[e], d = dst[e];
    // prefetch the next iteration's gather row (hide dependent-load latency)
    long t2 = t + stride;
    if (t2 < total) {
      long e2 = t2 / GROUPS;
      int g2 = (int)(t2 % GROUPS);
      int s2 = src[e2];
      __builtin_prefetch(h + (long)s2 * C + g2 * 4, 0, 0);  // global_prefetch_b8
    }
    float coef = dinv[s] * dinv[d];
    float4 hv = *(const float4*)(h + (long)s * C + g * 4);
    float* ap = agg + (long)d * C + g * 4;
    atomicAdd(ap + 0, hv.x * coef);
    atomicAdd(ap + 1, hv.y * coef);
    atomicAdd(ap + 2, hv.z * coef);
    atomicAdd(ap + 3, hv.w * coef);
  }
}

// agg[i,:] += h[i,:]*dinv[i]^2 + bias[:]   (self loop + conv bias), float4
template <int C>
__global__ void selfloop_bias_kernel(float4* __restrict__ agg, const float4* __restrict__ h,
                                     const float* __restrict__ dinv, const float* __restrict__ bias,
                                     long total4) {
  constexpr int CQ = C / 4;
  long i = (long)blockIdx.x * blockDim.x + threadIdx.x;
  long stride = (long)gridDim.x * blockDim.x;
  for (; i < total4; i += stride) {
    long r = i / CQ;
    int c0 = (int)(i % CQ) * 4;
    float di = dinv[r];
    float d2 = di * di;
    float4 av = agg[i], hv = h[i];
    av.x += hv.x * d2 + bias[c0 + 0];
    av.y += hv.y * d2 + bias[c0 + 1];
    av.z += hv.z * d2 + bias[c0 + 2];
    av.w += hv.w * d2 + bias[c0 + 3];
    agg[i] = av;
  }
}

// ------------------------------------------------------------- pair / tail

template <int C>
__global__ void pair_kernel(const float* __restrict__ xf, const int* __restrict__ pos,
                            float4* __restrict__ xp, long total4) {
  constexpr int CQ = C / 4;
  long i = (long)blockIdx.x * blockDim.x + threadIdx.x;
  long stride = (long)gridDim.x * blockDim.x;
  for (; i < total4; i += stride) {
    long p = i / CQ;
    int g = (int)(i % CQ);
    long r0 = pos[2 * p], r1 = pos[2 * p + 1];
    float4 a = *(const float4*)(xf + r0 * C + g * 4);
    float4 b = *(const float4*)(xf + r1 * C + g * 4);
    xp[i] = make_float4(a.x * b.x, a.y * b.y, a.z * b.z, a.w * b.w);
  }
}

__global__ void add_inplace4_kernel(float4* __restrict__ a, const float4* __restrict__ b, long n4) {
  long i = (long)blockIdx.x * blockDim.x + threadIdx.x;
  long stride = (long)gridDim.x * blockDim.x;
  for (; i < n4; i += stride) {
    float4 x = a[i], y = b[i];
    a[i] = make_float4(x.x + y.x, x.y + y.y, x.z + y.z, x.w + y.w);
  }
}

// out[i] = sum_c xs[idx[2i],c]*xs[idx[2i+1],c]*pw[c] + pb   (C=24 -> 6 float4)
__global__ void final_kernel(const float* __restrict__ xs, const int* __restrict__ idx,
                             const float* __restrict__ pw, const float* __restrict__ pb,
                             float* __restrict__ out, long R) {
  long i = (long)blockIdx.x * blockDim.x + threadIdx.x;
  long stride = (long)gridDim.x * blockDim.x;
  for (; i < R; i += stride) {
    long r0 = idx[2 * i], r1 = idx[2 * i + 1];
    const float4* p0 = (const float4*)(xs + r0 * 24);
    const float4* p1 = (const float4*)(xs + r1 * 24);
    const float4* pwv = (const float4*)pw;
    float acc = 0.0f;
#pragma unroll
    for (int q = 0; q < 6; ++q) {
      float4 a = p0[q], b = p1[q], w = pwv[q];
      acc += a.x * b.x * w.x + a.y * b.y * w.y + a.z * b.z * w.z + a.w * b.w * w.w;
    }
    out[i] = acc + pb[0];
  }
}

// ---------------------------------------------------------------- host side

static inline int gblocks(long n, int t) {
  long b = (n + t - 1) / t;
  if (b > 65535L * 16) b = 65535L * 16;
  if (b < 1) b = 1;
  return (int)b;
}

template <int C>
static void run_graphnorm(float* x, const float* w, const float* b, const float* a,
                          long Nn, int relu, float* stats, hipStream_t stream) {
  long total4 = Nn * C / 4;
  float invN = 1.0f / (float)Nn;
  fill4_kernel<<<1, 64, 0, stream>>>((float4*)stats, 0.0f, 16);
  colsum_kernel<C><<<gblocks(total4, 256), 256, 0, stream>>>((const float4*)x, a, nullptr,
                                                             stats, total4, invN, 0);
  colsum_kernel<C><<<gblocks(total4, 256), 256, 0, stream>>>((const float4*)x, a, stats,
                                                             stats + 32, total4, invN, 1);
  gnorm_apply_kernel<C><<<gblocks(total4, 256), 256, 0, stream>>>((float4*)x, stats, stats + 32,
                                                                  w, b, a, total4, invN, relu);
}

extern "C" void kernel_launch(void* const* d_in, const int* in_sizes, int n_in,
                              void* d_out, int out_size, void* d_ws, size_t ws_size,
                              hipStream_t stream) {
  (void)n_in; (void)ws_size;
  const int*   x_ids = (const int*)d_in[0];
  const int*   edge1 = (const int*)d_in[1];
  const int*   pos   = (const int*)d_in[2];
  const int*   ei2   = (const int*)d_in[3];
  const int*   idx   = (const int*)d_in[4];
  const float* emb   = (const float*)d_in[5];
  const float *gn0w = (const float*)d_in[6],  *gn0b = (const float*)d_in[7],  *gn0a = (const float*)d_in[8];
  const float *c1w0 = (const float*)d_in[9],  *c1b0 = (const float*)d_in[10];
  const float *g10w = (const float*)d_in[11], *g10b = (const float*)d_in[12], *g10a = (const float*)d_in[13];
  const float *c1w1 = (const float*)d_in[14], *c1b1 = (const float*)d_in[15];
  const float *g11w = (const float*)d_in[16], *g11b = (const float*)d_in[17], *g11a = (const float*)d_in[18];
  const float *c2w  = (const float*)d_in[19], *c2b  = (const float*)d_in[20];
  const float *g2w  = (const float*)d_in[21], *g2b  = (const float*)d_in[22], *g2a  = (const float*)d_in[23];
  const float *c2rw = (const float*)d_in[24], *c2rb = (const float*)d_in[25];
  const float *g2rw = (const float*)d_in[26], *g2rb = (const float*)d_in[27], *g2ra = (const float*)d_in[28];
  const float *pw   = (const float*)d_in[29], *pb   = (const float*)d_in[30];

  const long N  = in_sizes[0];
  const long E1 = in_sizes[1] / 2;
  const long P  = in_sizes[2] / 2;
  const long E2 = in_sizes[3] / 2;
  const int  C1 = 32, C2 = 24;
  const long Npad = (N + 63) & ~63L;     // GEMM tiles need no row guards
  const long Ppad = (P + 63) & ~63L;

  // workspace carve (floats); every carve is a multiple of 16B
  float* ws = (float*)d_ws;
  size_t off = 0;
  auto carve = [&](size_t n) { float* p = ws + off; off += n; return p; };
  float* bufA  = carve((size_t)Npad * C1);
  float* bufB  = carve((size_t)Npad * C1);
  float* bufC  = carve((size_t)Npad * C1);
  float* dinv1 = carve((size_t)Npad);
  float* dinv2a = carve((size_t)Ppad);
  float* dinv2b = carve((size_t)Ppad);
  float* stats = carve(64);
  float* xp   = carve((size_t)Ppad * C2);
  float* hp   = carve((size_t)Ppad * C2);
  float* aggA = carve((size_t)Ppad * C2);
  float* aggB = carve((size_t)Ppad * C2);

  const int* e1src = edge1;
  const int* e1dst = edge1 + E1;
  const int* e2src = ei2;
  const int* e2dst = ei2 + E2;

  const int SCAT_BLOCKS = 8192;   // grid-stride so the prefetch has work to hide

  // degrees / dinv for graph 1 (shared by both conv1 layers)
  fill4_kernel<<<gblocks(Npad / 4, 256), 256, 0, stream>>>((float4*)dinv1, 1.0f, Npad / 4);
  deg_kernel<<<gblocks(E1, 256), 256, 0, stream>>>(e1dst, dinv1, E1);
  rsqrt_kernel<<<gblocks(N, 256), 256, 0, stream>>>(dinv1, N);

  // x = GraphNorm(emb[x_ids])
  gather_emb_kernel<<<gblocks(N * 8, 256), 256, 0, stream>>>((const float4*)emb, x_ids,
                                                             (float4*)bufA, N * 8);
  run_graphnorm<32>(bufA, gn0w, gn0b, gn0a, N, 0, stats, stream);

  // conv1_0: h = x@W0 ; scatter + selfloop + bias ; GraphNorm ; ReLU
  gemm_wmma_kernel<32, 32><<<(int)(Npad / 64), 256, 0, stream>>>(bufA, c1w0, bufB);
  fill4_kernel<<<gblocks(N * 8, 256), 256, 0, stream>>>((float4*)bufC, 0.0f, N * 8);
  scatter_kernel<32><<<SCAT_BLOCKS, 256, 0, stream>>>(bufB, e1src, e1dst, dinv1, bufC, E1);
  selfloop_bias_kernel<32><<<gblocks(N * 8, 256), 256, 0, stream>>>((float4*)bufC, (const float4*)bufB,
                                                                    dinv1, c1b0, N * 8);
  run_graphnorm<32>(bufC, g10w, g10b, g10a, N, 1, stats, stream);

  // conv1_1: 32 -> 24, no activation
  gemm_wmma_kernel<32, 24><<<(int)(Npad / 64), 256, 0, stream>>>(bufC, c1w1, bufA);
  fill4_kernel<<<gblocks(N * 6, 256), 256, 0, stream>>>((float4*)bufB, 0.0f, N * 6);
  scatter_kernel<24><<<SCAT_BLOCKS, 256, 0, stream>>>(bufA, e1src, e1dst, dinv1, bufB, E1);
  selfloop_bias_kernel<24><<<gblocks(N * 6, 256), 256, 0, stream>>>((float4*)bufB, (const float4*)bufA,
                                                                    dinv1, c1b1, N * 6);
  run_graphnorm<24>(bufB, g11w, g11b, g11a, N, 0, stats, stream);

  // pair features: xp[p] = x[pos[p,0]] * x[pos[p,1]]
  pair_kernel<24><<<gblocks(P * 6, 256), 256, 0, stream>>>(bufB, pos, (float4*)xp, P * 6);

  // degrees for graph 2, forward and reversed
  fill4_kernel<<<gblocks(Ppad / 4, 256), 256, 0, stream>>>((float4*)dinv2a, 1.0f, Ppad / 4);
  deg_kernel<<<gblocks(E2, 256), 256, 0, stream>>>(e2dst, dinv2a, E2);
  rsqrt_kernel<<<gblocks(P, 256), 256, 0, stream>>>(dinv2a, P);
  fill4_kernel<<<gblocks(Ppad / 4, 256), 256, 0, stream>>>((float4*)dinv2b, 1.0f, Ppad / 4);
  deg_kernel<<<gblocks(E2, 256), 256, 0, stream>>>(e2src, dinv2b, E2);
  rsqrt_kernel<<<gblocks(P, 256), 256, 0, stream>>>(dinv2b, P);

  // conv2 (forward edges) -> GraphNorm -> ReLU
  gemm_wmma_kernel<24, 24><<<(int)(Ppad / 64), 256, 0, stream>>>(xp, c2w, hp);
  fill4_kernel<<<gblocks(P * 6, 256), 256, 0, stream>>>((float4*)aggA, 0.0f, P * 6);
  scatter_kernel<24><<<SCAT_BLOCKS, 256, 0, stream>>>(hp, e2src, e2dst, dinv2a, aggA, E2);
  selfloop_bias_kernel<24><<<gblocks(P * 6, 256), 256, 0, stream>>>((float4*)aggA, (const float4*)hp,
                                                                    dinv2a, c2b, P * 6);
  run_graphnorm<24>(aggA, g2w, g2b, g2a, P, 1, stats, stream);

  // conv2_r (reversed edges) -> GraphNorm -> ReLU
  gemm_wmma_kernel<24, 24><<<(int)(Ppad / 64), 256, 0, stream>>>(xp, c2rw, hp);
  fill4_kernel<<<gblocks(P * 6, 256), 256, 0, stream>>>((float4*)aggB, 0.0f, P * 6);
  scatter_kernel<24><<<SCAT_BLOCKS, 256, 0, stream>>>(hp, e2dst, e2src, dinv2b, aggB, E2);
  selfloop_bias_kernel<24><<<gblocks(P * 6, 256), 256, 0, stream>>>((float4*)aggB, (const float4*)hp,
                                                                    dinv2b, c2rb, P * 6);
  run_graphnorm<24>(aggB, g2rw, g2rb, g2ra, P, 1, stats, stream);

  // sum branches, gather idx, even*odd, dot with pw
  add_inplace4_kernel<<<gblocks(P * 6, 256), 256, 0, stream>>>((float4*)aggA, (const float4*)aggB, P * 6);
  final_kernel<<<gblocks(out_size, 256), 256, 0, stream>>>(aggA, idx, pw, pb,
                                                           (float*)d_out, out_size);
}